// Decoder_62740882260146
// MI455X (gfx1250) — compile-verified
//
#include <hip/hip_runtime.h>

typedef __attribute__((ext_vector_type(2))) float v2f;
typedef __attribute__((ext_vector_type(8))) float v8f;

#define B_     64
#define C_     4096
#define HID_   64
#define TFC_   48
#define THIST_ 24
#define WAVES  2      // waves per workgroup (16 rows each)
#define WPAD   68     // padded dword stride for W_hh in LDS (conflict-free b64 frags)
#define HPAD   68     // padded dword stride for h tile in LDS

__device__ __forceinline__ float sigm_(float x) { return 1.f / (1.f + __expf(-x)); }
__device__ __forceinline__ float tanh_(float x) { return 1.f - 2.f / (__expf(2.f * x) + 1.f); }

__device__ __forceinline__ v8f wmma4(v2f a, v2f b, v8f c) {
  // V_WMMA_F32_16X16X4_F32 : D = A(16x4,f32) * B(4x16,f32) + C(16x16,f32)
  return __builtin_amdgcn_wmma_f32_16x16x4_f32(false, a, false, b, (short)0, c, false, false);
}

// ---- prep: fold fc_in into W_ih ------------------------------------------
// W_comb[k][g] = sum_m W_ih[g][m] * W_in[m][k]         (ws[0 .. 767],  [4][192])
// b_comb[g]    = sum_m W_ih[g][m] * b_in[m] + b_ih[g]  (ws[768 .. 959])
__global__ void prep_fuse(const float* __restrict__ W_in, const float* __restrict__ b_in,
                          const float* __restrict__ W_ih, const float* __restrict__ b_ih,
                          float* __restrict__ ws) {
  int g = threadIdx.x;
  if (g < 192) {
    float acc0 = 0.f, acc1 = 0.f, acc2 = 0.f, acc3 = 0.f;
    float bacc = b_ih[g];
    for (int m = 0; m < HID_; ++m) {
      float w = W_ih[g * HID_ + m];
      acc0 += w * W_in[m * 4 + 0];
      acc1 += w * W_in[m * 4 + 1];
      acc2 += w * W_in[m * 4 + 2];
      acc3 += w * W_in[m * 4 + 3];
      bacc += w * b_in[m];
    }
    ws[0 * 192 + g] = acc0;
    ws[1 * 192 + g] = acc1;
    ws[2 * 192 + g] = acc2;
    ws[3 * 192 + g] = acc3;
    ws[768 + g]     = bacc;
  }
}

// ---- main GRU decoder -----------------------------------------------------
__global__ void __launch_bounds__(WAVES * 32, 4)
gru_decoder(const float* __restrict__ X,   const float* __restrict__ H,
            const float* __restrict__ xn,  const float* __restrict__ Whh,
            const float* __restrict__ bhh, const float* __restrict__ Wout,
            const float* __restrict__ bout,
            const float* __restrict__ Wcomb, const float* __restrict__ bcomb,
            float* __restrict__ out) {
  __shared__ float s_whh[192 * WPAD];        // W_hh row-major, padded
  __shared__ float s_h[WAVES][16 * HPAD];    // per-wave h tile (D-layout -> A-layout bridge)
  __shared__ float s_x[WAVES][16];           // per-wave recurrent scalar output

  const int tid   = threadIdx.x;
  const int lane  = tid & 31;
  const int wave  = tid >> 5;
  const int col16 = lane & 15;
  const int hi    = (lane >> 4) & 1;

  // stage W_hh (192x64) into padded LDS
  for (int idx = tid; idx < 192 * HID_; idx += WAVES * 32)
    s_whh[(idx >> 6) * WPAD + (idx & 63)] = Whh[idx];

  const int row0 = (blockIdx.x * WAVES + wave) * 16;  // 16 consecutive (b,c) rows
  const int b    = row0 >> 12;                         // C_ = 4096
  const int c0   = row0 & (C_ - 1);
  const int myc  = c0 + col16;

  // per-lane weight fragments (registers): fused-input B frags + bias scalars
  v2f   Bc[12];
  float cRZ[8], ciN[4], chN[4];
#pragma unroll
  for (int n = 0; n < 12; ++n) {
    int col  = col16 + 16 * n;
    Bc[n][0] = Wcomb[(hi ? 2 : 0) * 192 + col];
    Bc[n][1] = Wcomb[(hi ? 3 : 1) * 192 + col];
    float bc = bcomb[col];
    float bh = bhh[col];
    if (n < 8)      cRZ[n] = bc + bh;   // r/z: gi and gh share one accumulator
    else { ciN[n - 8] = bc; chN[n - 8] = bh; }
  }
  float wov[4];
#pragma unroll
  for (int j = 0; j < 4; ++j) wov[j] = Wout[col16 + 16 * j];
  const float b0 = bout[0];

  // h0 into WMMA C/D layout: 4 N-tiles of v8f
  v8f hacc[4];
#pragma unroll
  for (int j = 0; j < 4; ++j)
#pragma unroll
    for (int i = 0; i < 8; ++i)
      hacc[j][i] = H[(size_t)(row0 + i + hi * 8) * HID_ + col16 + 16 * j];

  if (lane < 16) s_x[wave][lane] = xn[row0 + lane];
  __syncthreads();

  const v8f zeroC = {0.f, 0.f, 0.f, 0.f, 0.f, 0.f, 0.f, 0.f};  // lowers to inline 0

  for (int t = 0; t < TFC_; ++t) {
    // ---- A fragment for fused fc_in GEMM: [x_prev, xt0, xt1, xt2] --------
    // Branchless: low lanes need {s_x, X+5}, high lanes need {X+6, X+7}.
    size_t xoff = ((size_t)(b * (THIST_ + TFC_) + THIST_ + t) * C_ + myc) * 8;
    float xg0 = X[xoff + 5 + hi];         // low: X5   high: X6
    float xg1 = X[xoff + 6 + hi];         // low: X6   high: X7
    float xprev = s_x[wave][col16];       // all lanes read (high lanes discard)
    v2f xA;
    xA[0] = hi ? xg0 : xprev;
    xA[1] = hi ? xg1 : xg0;

    // prefetch next step's X slice into near caches (consumed by this WGP)
    {
      int tn = (t + 1 < TFC_) ? (t + 1) : t;
      size_t xoffn = ((size_t)(b * (THIST_ + TFC_) + THIST_ + tn) * C_ + myc) * 8;
      __builtin_prefetch(&X[xoffn + 5], 0, 3);
    }

    // ---- gi (single K=4 WMMA per tile, inline-0 C) -----------------------
    v8f accR[4], accZ[4], accNi[4], accNh[4];
#pragma unroll
    for (int j = 0; j < 4; ++j) {
      accR[j]  = wmma4(xA, Bc[j],     zeroC);
      accZ[j]  = wmma4(xA, Bc[4 + j], zeroC);
      accNi[j] = wmma4(xA, Bc[8 + j], zeroC);
    }

    // ---- spill h (D-layout) into padded LDS for A-fragment extraction ----
#pragma unroll
    for (int j = 0; j < 4; ++j)
#pragma unroll
      for (int i = 0; i < 8; ++i)
        s_h[wave][(i + hi * 8) * HPAD + col16 + 16 * j] = hacc[j][i];

    // ---- gh = h @ W_hh^T : 16 K-slices x 12 N-tiles of 16x16x4 WMMA ------
    // Peel kk=0 so accNh is initialized by a WMMA with inline-0 C (no movs).
    {
      const float* hp = &s_h[wave][col16 * HPAD + hi * 2];
      v2f hA; hA[0] = hp[0]; hA[1] = hp[1];
#pragma unroll
      for (int j = 0; j < 4; ++j) {
        const float* wr = &s_whh[(col16 + 16 * (0 + j)) * WPAD + hi * 2];
        const float* wz = &s_whh[(col16 + 16 * (4 + j)) * WPAD + hi * 2];
        const float* wn = &s_whh[(col16 + 16 * (8 + j)) * WPAD + hi * 2];
        v2f fr; fr[0] = wr[0]; fr[1] = wr[1];
        v2f fz; fz[0] = wz[0]; fz[1] = wz[1];
        v2f fn; fn[0] = wn[0]; fn[1] = wn[1];
        accR[j]  = wmma4(hA, fr, accR[j]);
        accZ[j]  = wmma4(hA, fz, accZ[j]);
        accNh[j] = wmma4(hA, fn, zeroC);
      }
    }
#pragma unroll 2
    for (int kk = 1; kk < 16; ++kk) {
      const float* hp = &s_h[wave][col16 * HPAD + 4 * kk + hi * 2];
      v2f hA; hA[0] = hp[0]; hA[1] = hp[1];
#pragma unroll
      for (int j = 0; j < 4; ++j) {
        const float* wr = &s_whh[(col16 + 16 * (0 + j)) * WPAD + 4 * kk + hi * 2];
        const float* wz = &s_whh[(col16 + 16 * (4 + j)) * WPAD + 4 * kk + hi * 2];
        const float* wn = &s_whh[(col16 + 16 * (8 + j)) * WPAD + 4 * kk + hi * 2];
        v2f fr; fr[0] = wr[0]; fr[1] = wr[1];
        v2f fz; fz[0] = wz[0]; fz[1] = wz[1];
        v2f fn; fn[0] = wn[0]; fn[1] = wn[1];
        accR[j]  = wmma4(hA, fr, accR[j]);
        accZ[j]  = wmma4(hA, fz, accZ[j]);
        accNh[j] = wmma4(hA, fn, accNh[j]);
      }
    }

    // ---- gates (+ biases), state update, fc_out partial dot --------------
    float p[8];
#pragma unroll
    for (int i = 0; i < 8; ++i) p[i] = 0.f;
#pragma unroll
    for (int j = 0; j < 4; ++j)
#pragma unroll
      for (int i = 0; i < 8; ++i) {
        float r  = sigm_(accR[j][i] + cRZ[j]);
        float z  = sigm_(accZ[j][i] + cRZ[4 + j]);
        float n  = tanh_(accNi[j][i] + ciN[j] + r * (accNh[j][i] + chN[j]));
        float hn = (1.f - z) * n + z * hacc[j][i];
        hacc[j][i] = hn;
        p[i] += hn * wov[j];
      }

    // reduce across the 16 lanes of each half (each lane = one column)
#pragma unroll
    for (int i = 0; i < 8; ++i) {
      p[i] += __shfl_xor(p[i], 1, 16);
      p[i] += __shfl_xor(p[i], 2, 16);
      p[i] += __shfl_xor(p[i], 4, 16);
      p[i] += __shfl_xor(p[i], 8, 16);
    }
    if (lane == 0) {
#pragma unroll
      for (int i = 0; i < 8; ++i) s_x[wave][i] = p[i] + b0;       // rows 0..7
    }
    if (lane == 16) {
#pragma unroll
      for (int i = 0; i < 8; ++i) s_x[wave][8 + i] = p[i] + b0;   // rows 8..15
    }
    float xo = s_x[wave][col16];
    if (lane < 16)
      out[((size_t)b * TFC_ + t) * C_ + myc] = xo;
    // s_x now also feeds next step's x_prev (same-wave DS ordering)
  }
}

extern "C" void kernel_launch(void* const* d_in, const int* in_sizes, int n_in,
                              void* d_out, int out_size, void* d_ws, size_t ws_size,
                              hipStream_t stream) {
  (void)in_sizes; (void)n_in; (void)out_size; (void)ws_size;
  const float* X     = (const float*)d_in[0];
  const float* H     = (const float*)d_in[1];
  const float* xn    = (const float*)d_in[2];
  const float* W_in  = (const float*)d_in[3];
  const float* b_in  = (const float*)d_in[4];
  const float* W_ih  = (const float*)d_in[5];
  const float* W_hh  = (const float*)d_in[6];
  const float* b_ih  = (const float*)d_in[7];
  const float* b_hh  = (const float*)d_in[8];
  const float* W_out = (const float*)d_in[9];
  const float* b_out = (const float*)d_in[10];
  float* out = (float*)d_out;
  float* ws  = (float*)d_ws;   // [0..767] W_comb, [768..959] b_comb

  hipLaunchKernelGGL(prep_fuse, dim3(1), dim3(192), 0, stream, W_in, b_in, W_ih, b_ih, ws);

  const int nblocks = (B_ * C_) / (16 * WAVES);  // 262144 rows / 32 per block = 8192
  hipLaunchKernelGGL(gru_decoder, dim3(nblocks), dim3(WAVES * 32), 0, stream,
                     X, H, xn, W_hh, b_hh, W_out, b_out, ws, ws + 768, out);
}